// NCNPredictor_76467597738497
// MI455X (gfx1250) — compile-verified
//
#include <hip/hip_runtime.h>

// Problem constants from the reference
#define N_NODES 10000
#define D_CH    256
#define N_EDGES 8192
#define SPAD    10016   // padded node count for s-arrays in workspace

typedef float v2f __attribute__((ext_vector_type(2)));
typedef float v8f __attribute__((ext_vector_type(8)));

// ---------------------------------------------------------------------------
// Kernel 1: s0[n] = x[n,:] . Wxs[256:512],  s1[n] = x[n,:] . Wxs[512:768]
// via V_WMMA_F32_16X16X4_F32 (exact fp32 matrix FMA path on CDNA5).
// One wave (32 lanes) handles 16 nodes; K loop = 256/4 = 64 WMMA issues.
//
// A (16x4 f32, 2 VGPR): lanes 0-15 hold K={k0,k0+1}, lanes 16-31 K={k0+2,k0+3},
//                       M = lane & 15 (ISA 7.12.2 "32-bit A-Matrix 16x4").
// B (4x16 f32, 2 VGPR): mirrored mapping, N = lane & 15; columns 0/1 carry
//                       W1/W2, other columns forced to 0 via multiplicative
//                       mask (no divergence -> EXEC stays all-ones for WMMA).
// D (16x16 f32, 8 VGPR): vgpr r: lanes0-15 -> M=r, lanes16-31 -> M=r+8; N=lane.
// We only need columns N=0 (s0) and N=1 (s1).
// ---------------------------------------------------------------------------
__global__ __launch_bounds__(32) void ncn_gemv_wmma(
    const float* __restrict__ x,
    const float* __restrict__ Wxs,
    float* __restrict__ s0,
    float* __restrict__ s1)
{
    const int lane  = threadIdx.x;          // wave32: lane id
    const int base  = blockIdx.x * 16;      // 16 nodes per wave; 625 blocks
    const int m     = lane & 15;
    const int khalf = (lane >> 4) << 1;     // 0 or 2
    const float bmask = (m < 2) ? 1.0f : 0.0f;
    const int   wsel  = (m == 1) ? 256 : 0; // W1 at [256..), W2 at [512..)

    const float* xrow = x + (size_t)(base + m) * D_CH + khalf;
    const float* wrow = Wxs + D_CH + wsel + khalf;

    v8f c = {};
    for (int k0 = 0; k0 < D_CH; k0 += 4) {
        v2f a, b;
        a[0] = xrow[k0 + 0];
        a[1] = xrow[k0 + 1];
        b[0] = wrow[k0 + 0] * bmask;
        b[1] = wrow[k0 + 1] * bmask;
        c = __builtin_amdgcn_wmma_f32_16x16x4_f32(
                /*neg_a=*/false, a, /*neg_b=*/false, b,
                /*c_mod=*/(short)0, c, /*reuse_a=*/false, /*reuse_b=*/false);
    }

    // Scatter columns 0/1 of D to s0/s1 (divergence OK after last WMMA).
    if (lane == 0) {
        #pragma unroll
        for (int r = 0; r < 8; ++r) s0[base + r] = c[r];
    } else if (lane == 16) {
        #pragma unroll
        for (int r = 0; r < 8; ++r) s0[base + 8 + r] = c[r];
    } else if (lane == 1) {
        #pragma unroll
        for (int r = 0; r < 8; ++r) s1[base + r] = c[r];
    } else if (lane == 17) {
        #pragma unroll
        for (int r = 0; r < 8; ++r) s1[base + 8 + r] = c[r];
    }
}

// ---------------------------------------------------------------------------
// Kernel 2: one 256-thread block per target edge.
//   out[e] = W0 . (x[ti]*x[tj]) + sum_{n in cn0} s0[n] + sum_{n in cn1} s1[n] + b
// Adjacency rows streamed as aligned uint4 (bool = 1 byte, values 0x00/0x01;
// row length 10000 B = 625 * 16 B, so all loads are 16B aligned).
// ---------------------------------------------------------------------------
__global__ __launch_bounds__(256) void ncn_edge_kernel(
    const float* __restrict__ x,
    const unsigned char* __restrict__ adj01,
    const unsigned char* __restrict__ adj1,
    const int* __restrict__ tar_ei,
    const float* __restrict__ Wxs,
    const float* __restrict__ bxs,
    const float* __restrict__ s0,
    const float* __restrict__ s1,
    float* __restrict__ out)
{
    const int e  = blockIdx.x;
    const int t  = threadIdx.x;
    const int ti = tar_ei[e];
    const int tj = tar_ei[N_EDGES + e];

    // term: W0 . (x[ti] * x[tj]) -- blockDim == D, one element per thread
    float acc = x[(size_t)ti * D_CH + t] * x[(size_t)tj * D_CH + t] * Wxs[t];

    // common-neighbor masked sums over the node dimension
    const uint4* r01i = (const uint4*)(adj01 + (size_t)ti * N_NODES);
    const uint4* r01j = (const uint4*)(adj01 + (size_t)tj * N_NODES);
    const uint4* r1i  = (const uint4*)(adj1  + (size_t)ti * N_NODES);
    const uint4* r1j  = (const uint4*)(adj1  + (size_t)tj * N_NODES);

    const int NCHUNK = N_NODES / 16;  // 625 uint4 chunks per row
    for (int ch = t; ch < NCHUNK; ch += 256) {
        const uint4 a = r01i[ch];
        const uint4 b = r01j[ch];
        const uint4 p = r1i[ch];
        const uint4 q = r1j[ch];

        unsigned c1w[4], c0w[4];
        c1w[0] = p.x & q.x;  c1w[1] = p.y & q.y;
        c1w[2] = p.z & q.z;  c1w[3] = p.w & q.w;
        c0w[0] = (a.x & b.x) & ~c1w[0];
        c0w[1] = (a.y & b.y) & ~c1w[1];
        c0w[2] = (a.z & b.z) & ~c1w[2];
        c0w[3] = (a.w & b.w) & ~c1w[3];

        const unsigned any = c0w[0] | c0w[1] | c0w[2] | c0w[3]
                           | c1w[0] | c1w[1] | c1w[2] | c1w[3];
        if (any != 0u) {   // rare: ~N*p^2 common neighbors per edge
            const int nb = ch * 16;
            #pragma unroll
            for (int w = 0; w < 4; ++w) {
                #pragma unroll
                for (int by = 0; by < 4; ++by) {
                    const int n = nb + w * 4 + by;
                    if ((c0w[w] >> (8 * by)) & 1u) acc += s0[n];
                    if ((c1w[w] >> (8 * by)) & 1u) acc += s1[n];
                }
            }
        }
    }

    // block reduction
    __shared__ float red[256];
    red[t] = acc;
    __syncthreads();
    #pragma unroll
    for (int off = 128; off > 0; off >>= 1) {
        if (t < off) red[t] += red[t + off];
        __syncthreads();
    }
    if (t == 0) out[e] = red[0] + bxs[0];
}

// ---------------------------------------------------------------------------
// Host-side launcher
// Inputs (setup_inputs order): x, adj01, adj1, tar_ei, Wxs, bxs, NCN_mode
// ---------------------------------------------------------------------------
extern "C" void kernel_launch(void* const* d_in, const int* in_sizes, int n_in,
                              void* d_out, int out_size, void* d_ws, size_t ws_size,
                              hipStream_t stream)
{
    const float*         x      = (const float*)d_in[0];
    const unsigned char* adj01  = (const unsigned char*)d_in[1]; // bool = 1 byte
    const unsigned char* adj1   = (const unsigned char*)d_in[2]; // bool = 1 byte
    const int*           tar_ei = (const int*)d_in[3];
    const float*         Wxs    = (const float*)d_in[4];
    const float*         bxs    = (const float*)d_in[5];
    // d_in[6] = NCN_mode (always 0 branch)

    float* s0 = (float*)d_ws;          // N floats (padded)
    float* s1 = s0 + SPAD;             // N floats (padded); total ~80 KB

    // Phase 1: per-node projection scalars via fp32 WMMA (625 waves).
    ncn_gemv_wmma<<<N_NODES / 16, 32, 0, stream>>>(x, Wxs, s0, s1);

    // Phase 2: one workgroup per edge, bandwidth-bound adjacency scan.
    ncn_edge_kernel<<<N_EDGES, 256, 0, stream>>>(x, adj01, adj1, tar_ei,
                                                 Wxs, bxs, s0, s1,
                                                 (float*)d_out);
}